// CCDecoder_7919919693896
// MI455X (gfx1250) — compile-verified
//
#include <hip/hip_runtime.h>

#ifndef __has_builtin
#define __has_builtin(x) 0
#endif

#define DT      0.03f
#define SEQ     30
#define DSTEER  (0.4f * 0.03f)   /* STEER_RATE * DT */

typedef int v4i __attribute__((ext_vector_type(4)));
typedef __attribute__((address_space(1))) v4i  g_v4i;
typedef __attribute__((address_space(3))) v4i  l_v4i;
typedef __attribute__((address_space(3))) char l_char;

__device__ __forceinline__ void async_store_b128(void* gptr, void* lptr) {
#if __has_builtin(__builtin_amdgcn_global_store_async_from_lds_b128)
    __builtin_amdgcn_global_store_async_from_lds_b128(
        (g_v4i*)gptr, (l_v4i*)lptr, /*offset=*/0, /*cpol=*/0);
#else
    unsigned laddr = (unsigned)(unsigned long long)(l_char*)lptr;
    asm volatile("global_store_async_from_lds_b128 %0, %1, off"
                 :: "v"(gptr), "v"(laddr) : "memory");
#endif
}

__device__ __forceinline__ void wait_async_zero() {
#if __has_builtin(__builtin_amdgcn_s_wait_asynccnt)
    __builtin_amdgcn_s_wait_asynccnt(0);
#else
    asm volatile("s_wait_asynccnt 0" ::: "memory");
#endif
}

__global__ __launch_bounds__(128) void ccdecoder_rollout(
    const float* __restrict__ z,          // (n, 2)
    const float* __restrict__ init_state, // (n, 6)
    float* __restrict__ out,              // (n, 30, 4)
    int n)
{
    // [wave][step*32 + lane] float4 : conflict-free ds_store_b128 (16B lane stride)
    __shared__ float4 tile[4][SEQ * 32];

    const int lane = threadIdx.x & 31;
    const int w    = threadIdx.x >> 5;
    const int i    = blockIdx.x * 128 + threadIdx.x;

    if (i < n) {
        // ---- inputs (reads are tiny vs writes; plain b64 loads are fine) ----
        const float z0 = z[2 * i + 0];
        const float z1 = z[2 * i + 1];
        const float* st = init_state + (size_t)i * 6;
        const float2 p0 = *(const float2*)(st + 0);   // x, y
        const float2 p1 = *(const float2*)(st + 2);   // psi, v
        const float2 p2 = *(const float2*)(st + 4);   // _, last_st
        float x = p0.x, y = p0.y, psi = p1.x, v = p1.y;
        const float last_st = p2.y;

        // ---- steering / pedal preamble (clip_by_tensor quirk kept: t==t_min -> 0) ----
        const float steer0 = z1 * 0.5f;
        const float tmn = last_st - DSTEER;
        const float tmx = last_st + DSTEER;
        float r = (steer0 > tmn ? steer0 : 0.0f) + (steer0 < tmn ? tmn : 0.0f);
        r       = (r <= tmx   ? r      : 0.0f) + (r > tmx      ? tmx : 0.0f);
        const float beta = fminf(fmaxf(r, -0.5f), 0.5f);
        const float a_t  = fminf(fmaxf(z0 * 2.5f, -2.5f), 2.5f);
        const float kpsi = __tanf(beta) * (DT / 2.5f);   // tan(beta)/2.5*DT

        // ---- 30-step rollout, results staged into LDS ----
        float4* row = &tile[w][lane];
#pragma unroll
        for (int t = 0; t < SEQ; ++t) {
            const float v1 = fminf(fmaxf(fmaf(a_t, DT, v), 0.0f), 10.0f);
            psi = fmaf(v, kpsi, psi);                // uses v_t (pre-update)
            const float sp = __sinf(psi);
            const float cp = __cosf(psi);
            x = fmaf(v1 * DT, cp, x);
            y = fmaf(v1 * DT, sp, y);
            v = v1;
            row[t * 32] = make_float4(x, y, psi, v); // ds_store_b128, no bank conflicts
        }
    }

    // ---- stream the wave's 15360B tile to memory, transposed to element-major ----
    const int e0 = blockIdx.x * 128 + (w << 5);      // first element of this wave
    float4* wtile = &tile[w][0];

    if (e0 + 32 <= n) {
        // ensure the ds_stores above have landed before the async engine reads LDS
        asm volatile("s_wait_dscnt 0" ::: "memory");
        float4* gbase = (float4*)(out + (size_t)e0 * (SEQ * 4));
#pragma unroll
        for (int k = 0; k < SEQ; ++k) {
            const unsigned m  = (unsigned)(k * 32 + lane); // chunk index in wave region
            const unsigned lg = m / 30u;                   // element within wave
            const unsigned tg = m - lg * 30u;              // step
            // global side: 32 lanes write one contiguous 512B span (fully coalesced)
            async_store_b128(gbase + m, wtile + (tg * 32 + lg));
        }
        wait_async_zero();
    } else if (i < n) {
        // tail wave: per-thread copy of its own trajectory from LDS
        float4* gbase = (float4*)(out + (size_t)i * (SEQ * 4));
#pragma unroll
        for (int t = 0; t < SEQ; ++t)
            gbase[t] = wtile[t * 32 + lane];
    }
}

extern "C" void kernel_launch(void* const* d_in, const int* in_sizes, int n_in,
                              void* d_out, int out_size, void* d_ws, size_t ws_size,
                              hipStream_t stream) {
    const float* z          = (const float*)d_in[0];
    const float* init_state = (const float*)d_in[1];
    float* out              = (float*)d_out;
    const int n = in_sizes[0] / 2;                   // z is (n, 2)
    const int grid = (n + 127) / 128;
    ccdecoder_rollout<<<grid, 128, 0, stream>>>(z, init_state, out, n);
}